// clf_binaryv2_65704409694782
// MI455X (gfx1250) — compile-verified
//
#include <hip/hip_runtime.h>

typedef __attribute__((ext_vector_type(2))) float v2f;
typedef __attribute__((ext_vector_type(8))) float v8f;

#define GCN_N     100000
#define GCN_E     640000
#define GCN_DIM   128
#define GCN_LAYERS 3

// ---------------------------------------------------------------------------
// Degree / normalization
// ---------------------------------------------------------------------------
__global__ void deg_init_kernel(float* __restrict__ deg, int n) {
    int i = blockIdx.x * blockDim.x + threadIdx.x;
    if (i < n) deg[i] = 1.0f;               // self-loop contributes 1 to every node
}

__global__ void deg_count_kernel(const int* __restrict__ dst, float* __restrict__ deg, int e) {
    int i = blockIdx.x * blockDim.x + threadIdx.x;
    if (i < e) atomicAdd(&deg[dst[i]], 1.0f);
}

__global__ void dinv_kernel(float* __restrict__ deg, int n) {
    int i = blockIdx.x * blockDim.x + threadIdx.x;
    if (i < n) deg[i] = rsqrtf(deg[i]);     // deg >= 1 always (self-loops), matches where(deg>0)
}

// ---------------------------------------------------------------------------
// Y = (RELU ? relu(X) : X) @ W   — fp32 WMMA 16x16x4, templated on RELU so the
// inner loop is pure load+wmma (no cndmask chains).
// Block = 256 thr = 8 waves, covers 32 rows x 128 cols: wave w owns col tile w
// and BOTH row tiles (B fragment reused for 2 WMMAs). 100000/32 = 3125 exact.
// A frag (16x4): lanes 0-15 -> M=lane, K={k0,k0+1}; lanes 16-31 -> K={k0+2,k0+3}.
// B frag (4x16): lanes 0-15 -> N=lane, K={k0,k0+1}; lanes 16-31 -> K={k0+2,k0+3}.
// C/D: VGPR r -> M=r (lanes 0-15) and M=r+8 (lanes 16-31), N=lane&15.
// ---------------------------------------------------------------------------
template<bool RELU>
__global__ __launch_bounds__(256)
void gemm_wmma_f32_kernel(const float* __restrict__ X, const float* __restrict__ W,
                          float* __restrict__ Y) {
    const int lane = threadIdx.x & 31;
    const int wave = threadIdx.x >> 5;          // 0..7 : N tile
    const int half = lane >> 4;                 // 0 or 1
    const int l15  = lane & 15;
    const int row0 = blockIdx.x * 32;           // 2 M tiles per block
    const int col  = wave * 16 + l15;

    const float* Xr0 = X + (size_t)(row0 + l15) * GCN_DIM;
    const float* Xr1 = Xr0 + (size_t)16 * GCN_DIM;

    v8f acc0 = {}, acc1 = {};
    #pragma unroll 4
    for (int k0 = 0; k0 < GCN_DIM; k0 += 4) {
        const int ka = k0 + half * 2;
        v2f b;
        b.x = W[(size_t)(ka + 0) * GCN_DIM + col];
        b.y = W[(size_t)(ka + 1) * GCN_DIM + col];
        v2f a0, a1;
        a0.x = Xr0[ka]; a0.y = Xr0[ka + 1];
        a1.x = Xr1[ka]; a1.y = Xr1[ka + 1];
        if (RELU) {
            a0.x = fmaxf(a0.x, 0.0f); a0.y = fmaxf(a0.y, 0.0f);
            a1.x = fmaxf(a1.x, 0.0f); a1.y = fmaxf(a1.y, 0.0f);
        }
        acc0 = __builtin_amdgcn_wmma_f32_16x16x4_f32(
            false, a0, false, b, (short)0, acc0, false, false);
        acc1 = __builtin_amdgcn_wmma_f32_16x16x4_f32(
            false, a1, false, b, (short)0, acc1, false, false);
    }

    const int crow = row0 + half * 8;
    #pragma unroll
    for (int r = 0; r < 8; ++r) {
        Y[(size_t)(crow + r)      * GCN_DIM + col] = acc0[r];
        Y[(size_t)(crow + 16 + r) * GCN_DIM + col] = acc1[r];
    }
}

// ---------------------------------------------------------------------------
// acc[i][:] = xw[i][:] * dinv[i]^2 + bias     (self-loop message + bias, no atomics)
// one thread per float4 (N*32 threads)
// ---------------------------------------------------------------------------
__global__ __launch_bounds__(256)
void init_acc_kernel(const float4* __restrict__ xw, const float* __restrict__ dinv,
                     const float* __restrict__ bias, float4* __restrict__ acc) {
    int t = blockIdx.x * blockDim.x + threadIdx.x;
    if (t >= GCN_N * (GCN_DIM / 4)) return;
    int node = t >> 5;                          // 32 float4 groups per row
    int q    = t & 31;
    float s = dinv[node]; s = s * s;
    float4 v = xw[t];
    const float4 bb = reinterpret_cast<const float4*>(bias)[q];
    float4 r;
    r.x = v.x * s + bb.x; r.y = v.y * s + bb.y;
    r.z = v.z * s + bb.z; r.w = v.w * s + bb.w;
    acc[t] = r;
}

// ---------------------------------------------------------------------------
// Edge scatter: one wave32 per edge, lane owns 4 consecutive dims (float4).
// acc[dst] += xw[src] * dinv[src]*dinv[dst]   via global_atomic_add_f32
// ---------------------------------------------------------------------------
__global__ __launch_bounds__(256)
void scatter_kernel(const int* __restrict__ src, const int* __restrict__ dst,
                    const float* __restrict__ xw, const float* __restrict__ dinv,
                    float* __restrict__ acc) {
    int e = (blockIdx.x * blockDim.x + threadIdx.x) >> 5;
    if (e >= GCN_E) return;
    int lane = threadIdx.x & 31;
    int s = src[e];
    int d = dst[e];
    float norm = dinv[s] * dinv[d];
    float4 v = reinterpret_cast<const float4*>(xw + (size_t)s * GCN_DIM)[lane];
    float* ad = acc + (size_t)d * GCN_DIM + lane * 4;
    atomicAdd(ad + 0, v.x * norm);
    atomicAdd(ad + 1, v.y * norm);
    atomicAdd(ad + 2, v.z * norm);
    atomicAdd(ad + 3, v.w * norm);
}

// ---------------------------------------------------------------------------
// out[i][0:2] = relu(x[i]) @ clf_W + clf_b    — wave32 per node, shfl_xor reduce
// ---------------------------------------------------------------------------
__global__ __launch_bounds__(256)
void clf_kernel(const float* __restrict__ x, const float* __restrict__ cw,
                const float* __restrict__ cb, float* __restrict__ out) {
    int node = (blockIdx.x * blockDim.x + threadIdx.x) >> 5;
    if (node >= GCN_N) return;
    int lane = threadIdx.x & 31;
    float4 v = reinterpret_cast<const float4*>(x + (size_t)node * GCN_DIM)[lane];
    v.x = fmaxf(v.x, 0.0f); v.y = fmaxf(v.y, 0.0f);
    v.z = fmaxf(v.z, 0.0f); v.w = fmaxf(v.w, 0.0f);
    int d0 = lane * 4;
    float s0 = v.x * cw[(d0 + 0) * 2 + 0] + v.y * cw[(d0 + 1) * 2 + 0]
             + v.z * cw[(d0 + 2) * 2 + 0] + v.w * cw[(d0 + 3) * 2 + 0];
    float s1 = v.x * cw[(d0 + 0) * 2 + 1] + v.y * cw[(d0 + 1) * 2 + 1]
             + v.z * cw[(d0 + 2) * 2 + 1] + v.w * cw[(d0 + 3) * 2 + 1];
    #pragma unroll
    for (int off = 16; off >= 1; off >>= 1) {
        s0 += __shfl_xor(s0, off, 32);
        s1 += __shfl_xor(s1, off, 32);
    }
    if (lane == 0) {
        out[(size_t)node * 2 + 0] = s0 + cb[0];
        out[(size_t)node * 2 + 1] = s1 + cb[1];
    }
}

// ---------------------------------------------------------------------------
extern "C" void kernel_launch(void* const* d_in, const int* in_sizes, int n_in,
                              void* d_out, int out_size, void* d_ws, size_t ws_size,
                              hipStream_t stream) {
    (void)in_sizes; (void)n_in; (void)out_size; (void)ws_size;

    const float* nodes = (const float*)d_in[0];            // [N, 128]
    const int*   edges = (const int*)d_in[1];              // [2, E]
    const float* W     = (const float*)d_in[2];            // [3, 128, 128]
    const float* bias  = (const float*)d_in[3];            // [3, 128]
    const float* clf_W = (const float*)d_in[4];            // [128, 2]
    const float* clf_b = (const float*)d_in[5];            // [2]
    float*       out   = (float*)d_out;                    // [N, 2]

    const int* src = edges;
    const int* dst = edges + GCN_E;

    // workspace partition (256B-aligned)
    char* ws = (char*)d_ws;
    size_t off = 0;
    float* dinv = (float*)(ws + off); off += ((size_t)GCN_N * 4 + 255) & ~(size_t)255;
    float* xw   = (float*)(ws + off); off += (size_t)GCN_N * GCN_DIM * 4;
    float* acc  = (float*)(ws + off);

    const int T = 256;

    // degree -> dinv (in place)
    deg_init_kernel<<<(GCN_N + T - 1) / T, T, 0, stream>>>(dinv, GCN_N);
    deg_count_kernel<<<(GCN_E + T - 1) / T, T, 0, stream>>>(dst, dinv, GCN_E);
    dinv_kernel<<<(GCN_N + T - 1) / T, T, 0, stream>>>(dinv, GCN_N);

    const int gemm_blocks    = GCN_N / 32;                        // 3125, exact
    const int vec_threads    = GCN_N * (GCN_DIM / 4);             // N*32
    const int vec_blocks     = (vec_threads + T - 1) / T;
    const int scatter_blocks = (GCN_E * 32 + T - 1) / T;          // wave per edge
    const int clf_blocks     = (GCN_N * 32 + T - 1) / T;          // wave per node

    for (int layer = 0; layer < GCN_LAYERS; ++layer) {
        const float* xin = (layer == 0) ? nodes : acc;            // gemm reads acc before rewrite
        const float* Wl  = W + (size_t)layer * GCN_DIM * GCN_DIM;
        if (layer == 0)
            gemm_wmma_f32_kernel<false><<<gemm_blocks, T, 0, stream>>>(xin, Wl, xw);
        else
            gemm_wmma_f32_kernel<true><<<gemm_blocks, T, 0, stream>>>(xin, Wl, xw);
        init_acc_kernel<<<vec_blocks, T, 0, stream>>>(
            (const float4*)xw, dinv, bias + (size_t)layer * GCN_DIM, (float4*)acc);
        scatter_kernel<<<scatter_blocks, T, 0, stream>>>(src, dst, xw, dinv, acc);
    }

    clf_kernel<<<clf_blocks, T, 0, stream>>>(acc, clf_W, clf_b, out);
}